// FourierAttention_84464826843540
// MI455X (gfx1250) — compile-verified
//
#include <hip/hip_runtime.h>

// FourierAttention on MI455X (gfx1250):
//   out = LN(q) @ ( LN(k)^T @ v ) * 2^-14   (linear attention, associativity exploited)
// All matmuls on V_WMMA_F32_16X16X4_F32 (f32 to match the f32 reference).
// B-operands staged in a K-pair-interleaved LDS layout so every fragment is a
// single aligned ds_load_b64 (no VALU packing in the WMMA issue stream).

#define BATCH 4
#define SEQ   2048
#define DIM   512
#define HEADS 8
#define DK    64
#define NCOL  1536   // 3 * HEADS * DK
#define BPS   160    // paired-layout row stride in floats (160 % 64 == 32 -> bank-disjoint halves)

typedef __attribute__((ext_vector_type(2))) float v2f;
typedef __attribute__((ext_vector_type(8))) float v8f;

__device__ __forceinline__ v8f wmma4(v2f a, v2f b, v8f c) {
    // D = A(16x4 f32) * B(4x16 f32) + C(16x16 f32)
    return __builtin_amdgcn_wmma_f32_16x16x4_f32(
        /*neg_a=*/false, a, /*neg_b=*/false, b,
        /*c_mod=*/(short)0, c, /*reuse_a=*/false, /*reuse_b=*/false);
}

// Pack two adjacent K-rows (row stride = srcStride floats) at col pair 2*cp into
// the paired layout: dst[kp*BPS + 4*cp .. +3] = {r0.x, r1.x, r0.y, r1.y}
__device__ __forceinline__ void stage_pair(float* __restrict__ dst, int kp, int cp,
                                           const float* __restrict__ src, int srcStride) {
    const float* s = src + (size_t)(2 * kp) * srcStride + 2 * cp;
    float2 r0 = *(const float2*)s;
    float2 r1 = *(const float2*)(s + srcStride);
    float4 pk;
    pk.x = r0.x; pk.y = r1.x; pk.z = r0.y; pk.w = r1.y;
    *(float4*)&dst[kp * BPS + 4 * cp] = pk;
}

// ---------------------------------------------------------------------------
// Kernel 1: qkv = x @ W_qkv, scattered into head-split q/k/v  [B,H,N,64]
// grid = (24 col-blocks, 128 row-blocks), block = 256 (8 waves)
// ---------------------------------------------------------------------------
__global__ __launch_bounds__(256)
void qkv_gemm_kernel(const float* __restrict__ x, const float* __restrict__ w,
                     float* __restrict__ qb, float* __restrict__ kb,
                     float* __restrict__ vb) {
    __shared__ float lds_a[64][68];      // x tile  [token][k]
    __shared__ float lds_b[32 * BPS];    // W tile, K-pair interleaved: [kp][n][2]

    const int tid  = threadIdx.x;
    const int lane = tid & 31, wave = tid >> 5;
    const int half = lane >> 4, l16 = lane & 15;
    const int M0 = (wave & 3) * 16;          // token sub-block
    const int N0 = (wave >> 2) * 32;         // col sub-block (2 subtiles)

    const int cb    = blockIdx.x;            // 0..23 -> which of q/k/v + head
    const int col0  = cb * 64;
    const int which = cb >> 3, h = cb & 7;
    const int rowBase = blockIdx.y * 64;     // global token block
    float* dstBase = (which == 0) ? qb : (which == 1) ? kb : vb;

    const int lr  = tid >> 2;                // A-stage row 0..63
    const int lc  = (tid & 3) * 16;          // A-stage col quarter
    const int cp  = tid & 31;                // B-stage col pair 0..31
    const int kb8 = tid >> 5;                // B-stage kp base 0..7

    v8f acc0 = {}; v8f acc1 = {};

    for (int kc = 0; kc < DIM; kc += 64) {
        __syncthreads();
        #pragma unroll
        for (int i = 0; i < 4; ++i) {
            *(float4*)&lds_a[lr][lc + 4*i] =
                *(const float4*)&x[(size_t)(rowBase + lr) * DIM + kc + lc + 4*i];
            stage_pair(lds_b, kb8 + 8*i, cp, &w[(size_t)kc * NCOL + col0], NCOL);
        }
        __syncthreads();
        #pragma unroll
        for (int k0 = 0; k0 < 64; k0 += 4) {
            const int kp = (k0 >> 1) + half;
            v2f a  = *(const v2f*)&lds_a[M0 + l16][k0 + 2*half];
            v2f b0 = *(const v2f*)&lds_b[kp * BPS + 2*(N0 + l16)];
            v2f b1 = *(const v2f*)&lds_b[kp * BPS + 2*(N0 + 16 + l16)];
            acc0 = wmma4(a, b0, acc0);
            acc1 = wmma4(a, b1, acc1);
        }
    }

    #pragma unroll
    for (int r = 0; r < 8; ++r) {
        int rowg = rowBase + M0 + r + 8*half;       // global token
        int bb = rowg >> 11, n = rowg & (SEQ - 1);
        float* dst = dstBase + ((size_t)((bb * HEADS + h) * SEQ + n)) * DK;
        dst[N0 + l16]      = acc0[r];
        dst[N0 + 16 + l16] = acc1[r];
    }
}

// ---------------------------------------------------------------------------
// Kernel 2: per-head LayerNorm in place on [B,H,N,64]; one wave32 per row
// ---------------------------------------------------------------------------
__global__ __launch_bounds__(256)
void ln_kernel(float* __restrict__ t, const float* __restrict__ g,
               const float* __restrict__ bta) {
    const int row  = blockIdx.x * 8 + (threadIdx.x >> 5);
    const int lane = threadIdx.x & 31;
    const int h = (row >> 11) & (HEADS - 1);      // row = (b*H + h)*N + n

    float2 xv = *(float2*)(t + (size_t)row * DK + lane * 2);
    float s = xv.x + xv.y;
    #pragma unroll
    for (int o = 16; o >= 1; o >>= 1) s += __shfl_xor(s, o, 32);
    const float mu = s * (1.0f / DK);
    const float d0 = xv.x - mu, d1 = xv.y - mu;
    float v = d0 * d0 + d1 * d1;
    #pragma unroll
    for (int o = 16; o >= 1; o >>= 1) v += __shfl_xor(v, o, 32);
    const float rinv = rsqrtf(v * (1.0f / DK) + 1e-5f);

    float2 gv = *(const float2*)(g   + h * DK + lane * 2);
    float2 bv = *(const float2*)(bta + h * DK + lane * 2);
    float2 ov;
    ov.x = d0 * rinv * gv.x + bv.x;
    ov.y = d1 * rinv * gv.y + bv.y;
    *(float2*)(t + (size_t)row * DK + lane * 2) = ov;
}

// ---------------------------------------------------------------------------
// Kernel 3: KV[b,h] = (k^T @ v) * 2^-14   (64x64, K = SEQ), grid = 32 blocks
// Both k and v staged K(token)-pair interleaved: fragments are pure b64 loads.
// ---------------------------------------------------------------------------
__global__ __launch_bounds__(256)
void kv_gemm_kernel(const float* __restrict__ kbuf, const float* __restrict__ vbuf,
                    float* __restrict__ kvb) {
    __shared__ float lds_k[32 * BPS];   // [tp][d][2] : lds_k[tp*BPS+2d+z] = k[2tp+z][d]
    __shared__ float lds_v[32 * BPS];

    const int tid  = threadIdx.x;
    const int lane = tid & 31, wave = tid >> 5;
    const int half = lane >> 4, l16 = lane & 15;
    const int M0 = (wave & 3) * 16, N0 = (wave >> 2) * 32;
    const int bh = blockIdx.x;
    const float* kp_ptr = kbuf + (size_t)bh * SEQ * DK;
    const float* vp_ptr = vbuf + (size_t)bh * SEQ * DK;
    const int cp  = tid & 31;            // d pair
    const int tb8 = tid >> 5;            // tp base

    v8f acc0 = {}; v8f acc1 = {};

    for (int t0 = 0; t0 < SEQ; t0 += 64) {
        __syncthreads();
        #pragma unroll
        for (int i = 0; i < 4; ++i) {
            stage_pair(lds_k, tb8 + 8*i, cp, kp_ptr + (size_t)t0 * DK, DK);
            stage_pair(lds_v, tb8 + 8*i, cp, vp_ptr + (size_t)t0 * DK, DK);
        }
        __syncthreads();
        #pragma unroll
        for (int k0 = 0; k0 < 64; k0 += 4) {
            const int kp = (k0 >> 1) + half;
            // A = k^T : a = {k[tok][d], k[tok+1][d]} with tok = k0+2*half, d = M0+l16
            v2f a  = *(const v2f*)&lds_k[kp * BPS + 2*(M0 + l16)];
            v2f b0 = *(const v2f*)&lds_v[kp * BPS + 2*(N0 + l16)];
            v2f b1 = *(const v2f*)&lds_v[kp * BPS + 2*(N0 + 16 + l16)];
            acc0 = wmma4(a, b0, acc0);
            acc1 = wmma4(a, b1, acc1);
        }
    }

    const float scale = 1.0f / 16384.0f;   // DK^-0.5 / SEQ = (1/8)/2048, exact pow2
    float* kvd = kvb + (size_t)bh * DK * DK;
    #pragma unroll
    for (int r = 0; r < 8; ++r) {
        int m = M0 + r + 8*half;
        kvd[m * DK + N0 + l16]      = acc0[r] * scale;
        kvd[m * DK + N0 + 16 + l16] = acc1[r] * scale;
    }
}

// ---------------------------------------------------------------------------
// Kernel 4: out[b, n, h*64+d] = q[b,h,n,:] @ KV[b,h]   (K = 64)
// grid = (SEQ/64, B*H)
// ---------------------------------------------------------------------------
__global__ __launch_bounds__(256)
void out_gemm_kernel(const float* __restrict__ qbuf, const float* __restrict__ kvb,
                     float* __restrict__ out) {
    __shared__ float lds_q[64][68];      // q tile [token][d]
    __shared__ float lds_kv[32 * BPS];   // KV, K-pair interleaved

    const int tid  = threadIdx.x;
    const int lane = tid & 31, wave = tid >> 5;
    const int half = lane >> 4, l16 = lane & 15;
    const int M0 = (wave & 3) * 16, N0 = (wave >> 2) * 32;
    const int tok0 = blockIdx.x * 64;
    const int bh = blockIdx.y;
    const int bb = bh >> 3, h = bh & 7;
    const float* qp  = qbuf + (size_t)bh * SEQ * DK + (size_t)tok0 * DK;
    const float* kvp = kvb + (size_t)bh * DK * DK;
    const int lr  = tid >> 2, lc = (tid & 3) * 16;
    const int cp  = tid & 31, kb8 = tid >> 5;

    #pragma unroll
    for (int i = 0; i < 4; ++i) {
        *(float4*)&lds_q[lr][lc + 4*i] = *(const float4*)&qp[(size_t)lr * DK + lc + 4*i];
        stage_pair(lds_kv, kb8 + 8*i, cp, kvp, DK);
    }
    __syncthreads();

    v8f acc0 = {}; v8f acc1 = {};
    #pragma unroll
    for (int k0 = 0; k0 < 64; k0 += 4) {
        const int kp = (k0 >> 1) + half;
        v2f a  = *(const v2f*)&lds_q[M0 + l16][k0 + 2*half];   // A[M=tok][K=d]
        v2f b0 = *(const v2f*)&lds_kv[kp * BPS + 2*(N0 + l16)];
        v2f b1 = *(const v2f*)&lds_kv[kp * BPS + 2*(N0 + 16 + l16)];
        acc0 = wmma4(a, b0, acc0);
        acc1 = wmma4(a, b1, acc1);
    }

    #pragma unroll
    for (int r = 0; r < 8; ++r) {
        int n = tok0 + M0 + r + 8*half;
        float* dst = out + ((size_t)(bb * SEQ + n)) * (HEADS * DK) + h * DK;
        dst[N0 + l16]      = acc0[r];
        dst[N0 + 16 + l16] = acc1[r];
    }
}

// ---------------------------------------------------------------------------
extern "C" void kernel_launch(void* const* d_in, const int* in_sizes, int n_in,
                              void* d_out, int out_size, void* d_ws, size_t ws_size,
                              hipStream_t stream) {
    const float* x  = (const float*)d_in[0];   // [4,2048,512]
    const float* w  = (const float*)d_in[1];   // [512,1536]
    const float* gq = (const float*)d_in[2];   // [8,64]
    const float* bq = (const float*)d_in[3];
    const float* gk = (const float*)d_in[4];
    const float* bk = (const float*)d_in[5];
    float* out = (float*)d_out;                // [4,2048,512]

    float* ws = (float*)d_ws;
    const size_t perBuf = (size_t)BATCH * HEADS * SEQ * DK;  // 4,194,304 floats each
    float* qb  = ws;                  // q  [B,H,N,64]
    float* kb  = ws + perBuf;         // k  [B,H,N,64]
    float* vb  = ws + 2 * perBuf;     // v  [B,H,N,64]
    float* kvb = ws + 3 * perBuf;     // KV [B*H,64,64]   (total ~50.9 MB workspace)

    qkv_gemm_kernel<<<dim3(NCOL / 64, (BATCH * SEQ) / 64), 256, 0, stream>>>(x, w, qb, kb, vb);
    ln_kernel<<<dim3((BATCH * HEADS * SEQ) / 8), 256, 0, stream>>>(qb, gq, bq);
    ln_kernel<<<dim3((BATCH * HEADS * SEQ) / 8), 256, 0, stream>>>(kb, gk, bk);
    kv_gemm_kernel<<<dim3(BATCH * HEADS), 256, 0, stream>>>(kb, vb, kvb);
    out_gemm_kernel<<<dim3(SEQ / 64, BATCH * HEADS), 256, 0, stream>>>(qb, kvb, out);
}